// LRU_34857954574984
// MI455X (gfx1250) — compile-verified
//
#include <hip/hip_runtime.h>

typedef __attribute__((ext_vector_type(2))) float v2f;
typedef __attribute__((ext_vector_type(8))) float v8f;
typedef __attribute__((ext_vector_type(4))) int   v4i;

#define BATCH   8
#define SEQLEN  8192
#define H_IN    128
#define H_OUT   128
#define NSTATE  256
#define ROWS    (BATCH*SEQLEN)     // 65536
#define NCAT    512                // [re | im] outputs of gemm1
#define KCAT    640                // [xs_re(256) | xs_im(256) | u(128)]
#define CHUNK   256
#define NCHUNK  (SEQLEN/CHUNK)     // 32

#define GLOBAL_AS __attribute__((address_space(1)))
#define LDS_AS    __attribute__((address_space(3)))

// Async global->LDS (CDNA5 GLOBAL_LOAD_ASYNC_TO_LDS_B128, ASYNCcnt-tracked).
// Builtin signature (from hipcc diagnostic): (AS1 v4i*, AS3 v4i*, imm offset, imm cpol).
#if defined(__gfx1250__) && __has_builtin(__builtin_amdgcn_global_load_async_to_lds_b128)
#define HAVE_ASYNC 1
__device__ __forceinline__ void async_ld16(const float* g, float* l) {
    __builtin_amdgcn_global_load_async_to_lds_b128(
        (GLOBAL_AS v4i*)g, (LDS_AS v4i*)l, 0, 0);
}
__device__ __forceinline__ void wait_async0() {
#if __has_builtin(__builtin_amdgcn_s_wait_asynccnt)
    __builtin_amdgcn_s_wait_asynccnt(0);
#else
    asm volatile("s_wait_asynccnt 0x0" ::: "memory");
#endif
}
#endif

// ---------------------------------------------------------------------------
// K0: precompute lambda, lambda^CHUNK, fused weight matrices
// lam layout: [0..255]=lam_re, [256..511]=lam_im, [512..767]=lamT_re, [768..1023]=lamT_im
// ---------------------------------------------------------------------------
__global__ void lru_prep(const float* __restrict__ nu_log, const float* __restrict__ theta_log,
                         const float* __restrict__ gamma_log,
                         const float* __restrict__ Bp_re, const float* __restrict__ Bp_im,
                         const float* __restrict__ C_re, const float* __restrict__ C_im,
                         const float* __restrict__ Dm,
                         float* __restrict__ bcat, float* __restrict__ wcat,
                         float* __restrict__ lam) {
    int idx = blockIdx.x * blockDim.x + threadIdx.x;
    if (idx < NSTATE) {
        float mag = expf(-expf(nu_log[idx]));
        float th  = expf(theta_log[idx]);
        float lr = mag * cosf(th);
        float li = mag * sinf(th);
        lam[idx]       = lr;
        lam[256 + idx] = li;
        // lambda^CHUNK by repeated squaring (CHUNK = 2^8)
        float tr = lr, ti = li;
        #pragma unroll
        for (int s = 0; s < 8; ++s) {
            float nr = tr * tr - ti * ti;
            float ni = 2.0f * tr * ti;
            tr = nr; ti = ni;
        }
        lam[512 + idx] = tr;
        lam[768 + idx] = ti;
    }
    if (idx < NCAT * H_IN) {                 // 65536: Bcat[j][h] = gamma[j%256] * Bp_{re/im}
        int j = idx >> 7, h = idx & 127;
        int s = j & 255;
        float g = expf(gamma_log[s]);
        float v = (j < NSTATE) ? Bp_re[s * H_IN + h] : Bp_im[s * H_IN + h];
        bcat[idx] = g * v;
    }
    if (idx < H_OUT * KCAT) {                // 81920: Wcat[n][k] = [C_re | -C_im | D]
        int n = idx / KCAT, k = idx - n * KCAT;
        float v;
        if (k < NSTATE)          v =  C_re[n * NSTATE + k];
        else if (k < 2 * NSTATE) v = -C_im[n * NSTATE + (k - NSTATE)];
        else                     v =  Dm[n * H_IN + (k - 2 * NSTATE)];
        wcat[idx] = v;
    }
}

// ---------------------------------------------------------------------------
// K1: Bu = u(65536x128) @ Bcat^T(128x512) -> planar xs_re / xs_im (65536x256 each)
// Block = 256 thr (8 waves). M-tile = 16, each wave covers 4 N-tiles of 16.
// A tile in LDS (stride 132 -> conflict-free fragment reads), A frags in regs.
// A tile staged with async global->LDS B128 copies when available.
// ---------------------------------------------------------------------------
__global__ __launch_bounds__(256) void lru_gemm1(const float* __restrict__ u,
                                                 const float* __restrict__ bcat,
                                                 float* __restrict__ xs_re,
                                                 float* __restrict__ xs_im) {
    __shared__ __align__(16) float As[16 * 132];
    const int tid  = threadIdx.x;
    const int m0   = blockIdx.x * 16;
    const int lane = tid & 31;
    const int w    = tid >> 5;

#ifdef HAVE_ASYNC
    {
        const int laneoff = lane * 4;            // 4 floats = 16B per lane
        #pragma unroll
        for (int rr = 0; rr < 2; ++rr) {
            int r = w * 2 + rr;                  // each wave stages 2 rows
            async_ld16(&u[(m0 + r) * H_IN + laneoff], &As[r * 132 + laneoff]);
        }
        wait_async0();
    }
#else
    #pragma unroll
    for (int i = 0; i < 8; ++i) {
        int idx = tid + i * 256;
        int r = idx >> 7, c = idx & 127;
        As[r * 132 + c] = u[(m0 + r) * H_IN + c];
    }
#endif
    __syncthreads();

    const int arow = lane & 15;
    const int kp   = (lane >> 4) << 1;      // 0 or 2: K-pair select per ISA A layout

    v2f af[32];
    #pragma unroll
    for (int kk = 0; kk < 32; ++kk)
        af[kk] = *(const v2f*)&As[arow * 132 + kk * 4 + kp];

    #pragma unroll
    for (int j = 0; j < 4; ++j) {
        const int n0 = (w * 4 + j) * 16;
        const float* bptr = &bcat[(n0 + arow) * H_IN + kp];
        v8f acc = {};
        #pragma unroll
        for (int kk = 0; kk < 32; ++kk) {
            v2f bf = *(const v2f*)&bptr[kk * 4];
            acc = __builtin_amdgcn_wmma_f32_16x16x4_f32(
                false, af[kk], false, bf, (short)0, acc, false, false);
        }
        float* plane = (n0 < NSTATE) ? xs_re : xs_im;
        const int pc    = (n0 & 255) + arow;
        const int rbase = m0 + ((lane >> 4) << 3);
        #pragma unroll
        for (int v = 0; v < 8; ++v)
            plane[(rbase + v) * NSTATE + pc] = acc[v];
    }
}

// ---------------------------------------------------------------------------
// K2: per-chunk local scan, in place (Bu -> local xs). 65536 threads: (b, chunk, n)
// ---------------------------------------------------------------------------
__global__ void lru_scan_local(float* __restrict__ xs_re, float* __restrict__ xs_im,
                               const float* __restrict__ lam,
                               float* __restrict__ cend_re, float* __restrict__ cend_im) {
    int tid = blockIdx.x * blockDim.x + threadIdx.x;
    int n = tid & 255;
    int c = (tid >> 8) & (NCHUNK - 1);
    int b = tid >> 13;
    float lr = lam[n], li = lam[256 + n];
    int base = (b * SEQLEN + c * CHUNK) * NSTATE + n;
    float xr = 0.f, xi = 0.f;
    for (int t = 0; t < CHUNK; ++t) {
        int idx = base + t * NSTATE;
        float br = xs_re[idx], bi = xs_im[idx];
        float nr = fmaf(lr, xr, fmaf(-li, xi, br));
        float ni = fmaf(lr, xi, fmaf( li, xr, bi));
        xr = nr; xi = ni;
        xs_re[idx] = xr; xs_im[idx] = xi;
    }
    int ci = (b * NCHUNK + c) * NSTATE + n;
    cend_re[ci] = xr; cend_im[ci] = xi;
}

// ---------------------------------------------------------------------------
// K3: cross-chunk carry scan. 2048 threads: (b, n), 32 sequential chunks.
// cin[c] = exclusive carry into chunk c:  cin[c+1] = lam^CHUNK * cin[c] + cend[c]
// ---------------------------------------------------------------------------
__global__ void lru_scan_carry(const float* __restrict__ cend_re, const float* __restrict__ cend_im,
                               float* __restrict__ cin_re, float* __restrict__ cin_im,
                               const float* __restrict__ lam) {
    int tid = blockIdx.x * blockDim.x + threadIdx.x;
    int n = tid & 255;
    int b = tid >> 8;
    float Lr = lam[512 + n], Li = lam[768 + n];
    float xr = 0.f, xi = 0.f;
    for (int c = 0; c < NCHUNK; ++c) {
        int i = (b * NCHUNK + c) * NSTATE + n;
        cin_re[i] = xr; cin_im[i] = xi;
        float nr = fmaf(Lr, xr, fmaf(-Li, xi, cend_re[i]));
        float ni = fmaf(Lr, xi, fmaf( Li, xr, cend_im[i]));
        xr = nr; xi = ni;
    }
}

// ---------------------------------------------------------------------------
// K4: fixup xs += lam^(t+1) * cin, write planar xs back, and emit `states`
// (complex64 interleaved) to d_out, including the zero row at l==0.
// ---------------------------------------------------------------------------
__global__ void lru_fixup(float* __restrict__ xs_re, float* __restrict__ xs_im,
                          const float* __restrict__ cin_re, const float* __restrict__ cin_im,
                          const float* __restrict__ lam, float* __restrict__ states) {
    int tid = blockIdx.x * blockDim.x + threadIdx.x;
    int n = tid & 255;
    int c = (tid >> 8) & (NCHUNK - 1);
    int b = tid >> 13;
    float lr = lam[n], li = lam[256 + n];
    int ci = (b * NCHUNK + c) * NSTATE + n;
    float cr = cin_re[ci], cm = cin_im[ci];
    if (c == 0) {
        long s0 = ((long)b * (SEQLEN + 1)) * NSTATE + n;
        states[s0 * 2] = 0.f; states[s0 * 2 + 1] = 0.f;
    }
    float pr = lr, pi = li;                   // lam^(t+1)
    int base = (b * SEQLEN + c * CHUNK) * NSTATE + n;
    for (int t = 0; t < CHUNK; ++t) {
        int idx = base + t * NSTATE;
        float xr = xs_re[idx] + (pr * cr - pi * cm);
        float xi = xs_im[idx] + (pr * cm + pi * cr);
        xs_re[idx] = xr; xs_im[idx] = xi;
        long l  = (long)c * CHUNK + t;
        long so = (((long)b * (SEQLEN + 1)) + l + 1) * NSTATE + n;
        states[so * 2] = xr; states[so * 2 + 1] = xi;
        float npr = pr * lr - pi * li;
        float npi = pr * li + pi * lr;
        pr = npr; pi = npi;
    }
}

// ---------------------------------------------------------------------------
// K5: y = [xs(l-1)_re | xs(l-1)_im | u] (65536x640) @ Wcat^T(640x128)
// M-tile = 16, 8 waves each own one N-tile of 16, K staged in LDS (stride 644).
// The l-1 shift (and zero at l==0) handled at A-tile staging; staging uses
// async global->LDS B128 copies when available (5 issues/row, zero-fill l==0).
// ---------------------------------------------------------------------------
__global__ __launch_bounds__(256) void lru_gemm2(const float* __restrict__ u,
                                                 const float* __restrict__ xs_re,
                                                 const float* __restrict__ xs_im,
                                                 const float* __restrict__ wcat,
                                                 float* __restrict__ y) {
    __shared__ __align__(16) float As[16 * 644];
    const int tid  = threadIdx.x;
    const int m0   = blockIdx.x * 16;
    const int lane = tid & 31;
    const int w    = tid >> 5;

#ifdef HAVE_ASYNC
    {
        const int laneoff = lane * 4;            // 16B per lane
        #pragma unroll
        for (int rr = 0; rr < 2; ++rr) {
            int r  = w * 2 + rr;                 // each wave stages 2 rows
            int gr = m0 + r;
            int l  = gr & (SEQLEN - 1);
            float* dst = &As[r * 644];
            if (l == 0) {
                // first row of a batch: pre-state is zero
                #pragma unroll
                for (int c4 = 0; c4 < 4; ++c4)
                    *(float4*)&dst[(lane + c4 * 32) * 4] = make_float4(0.f, 0.f, 0.f, 0.f);
            } else {
                const float* sre = &xs_re[(gr - 1) * NSTATE];
                const float* sim = &xs_im[(gr - 1) * NSTATE];
                async_ld16(sre + laneoff,       dst + laneoff);
                async_ld16(sre + 128 + laneoff, dst + 128 + laneoff);
                async_ld16(sim + laneoff,       dst + 256 + laneoff);
                async_ld16(sim + 128 + laneoff, dst + 384 + laneoff);
            }
            async_ld16(&u[gr * H_IN] + laneoff, dst + 512 + laneoff);
        }
        wait_async0();
    }
#else
    for (int idx = tid; idx < 16 * KCAT; idx += 256) {
        int r = idx / KCAT, k = idx - r * KCAT;
        int gr = m0 + r;
        int l  = gr & (SEQLEN - 1);
        float v;
        if (k < 2 * NSTATE) {
            if (l == 0) v = 0.f;
            else {
                const float* p = (k < NSTATE) ? xs_re : xs_im;
                v = p[(gr - 1) * NSTATE + (k & 255)];
            }
        } else {
            v = u[gr * H_IN + (k - 2 * NSTATE)];
        }
        As[r * 644 + k] = v;
    }
#endif
    __syncthreads();

    const int arow = lane & 15;
    const int kp   = (lane >> 4) << 1;
    const int n0   = w * 16;
    const float* bptr = &wcat[(n0 + arow) * KCAT + kp];
    const float* aptr = &As[arow * 644 + kp];
    v8f acc = {};
    #pragma unroll 8
    for (int kk = 0; kk < KCAT / 4; ++kk) {
        v2f af = *(const v2f*)&aptr[kk * 4];
        v2f bf = *(const v2f*)&bptr[kk * 4];
        acc = __builtin_amdgcn_wmma_f32_16x16x4_f32(
            false, af, false, bf, (short)0, acc, false, false);
    }
    const int col   = n0 + arow;
    const int rbase = m0 + ((lane >> 4) << 3);
    #pragma unroll
    for (int v = 0; v < 8; ++v)
        y[(rbase + v) * H_OUT + col] = acc[v];
}

// ---------------------------------------------------------------------------
// Launch. Inputs: u, nu_log, theta_log, gamma_log, Bp_re, Bp_im, C_re, C_im, D
// Output: [ y (8*8192*128 f32) | states (8*8193*256 complex64, interleaved) ]
// Workspace: ~136 MB: xs_re, xs_im planes + fused weights + lam + carries.
// ---------------------------------------------------------------------------
extern "C" void kernel_launch(void* const* d_in, const int* in_sizes, int n_in,
                              void* d_out, int out_size, void* d_ws, size_t ws_size,
                              hipStream_t stream) {
    const float* u         = (const float*)d_in[0];
    const float* nu_log    = (const float*)d_in[1];
    const float* theta_log = (const float*)d_in[2];
    const float* gamma_log = (const float*)d_in[3];
    const float* Bp_re     = (const float*)d_in[4];
    const float* Bp_im     = (const float*)d_in[5];
    const float* C_re      = (const float*)d_in[6];
    const float* C_im      = (const float*)d_in[7];
    const float* Dm        = (const float*)d_in[8];

    float* ws      = (float*)d_ws;
    float* xs_re   = ws;
    float* xs_im   = ws + (size_t)ROWS * NSTATE;
    float* bcat    = ws + 2ull * ROWS * NSTATE;
    float* wcat    = bcat + (size_t)NCAT * H_IN;
    float* lam     = wcat + (size_t)H_OUT * KCAT;
    float* cend_re = lam + 1024;
    float* cend_im = cend_re + BATCH * NCHUNK * NSTATE;
    float* cin_re  = cend_im + BATCH * NCHUNK * NSTATE;
    float* cin_im  = cin_re  + BATCH * NCHUNK * NSTATE;

    float* y      = (float*)d_out;
    float* states = y + (size_t)ROWS * H_OUT;

    lru_prep<<<(H_OUT * KCAT + 255) / 256, 256, 0, stream>>>(
        nu_log, theta_log, gamma_log, Bp_re, Bp_im, C_re, C_im, Dm, bcat, wcat, lam);
    lru_gemm1<<<ROWS / 16, 256, 0, stream>>>(u, bcat, xs_re, xs_im);
    lru_scan_local<<<(BATCH * NCHUNK * NSTATE) / 256, 256, 0, stream>>>(
        xs_re, xs_im, lam, cend_re, cend_im);
    lru_scan_carry<<<(BATCH * NSTATE) / 256, 256, 0, stream>>>(
        cend_re, cend_im, cin_re, cin_im, lam);
    lru_fixup<<<(BATCH * NCHUNK * NSTATE) / 256, 256, 0, stream>>>(
        xs_re, xs_im, cin_re, cin_im, lam, states);
    lru_gemm2<<<ROWS / 16, 256, 0, stream>>>(u, xs_re, xs_im, wcat, y);
}